// GDKD_2353642078346
// MI455X (gfx1250) — compile-verified
//
#include <hip/hip_runtime.h>
#include <math.h>

// GDKD: y_s, y_t are [8, 19, 512, 512] f32. N = 8*512*512 pixels, C = 19.
// W0=1, W1=1, W2=2, T=1. Output: single f32 scalar.

#define CC 19
#define LOG2_HW 18
#define HW (1 << LOG2_HW)   // 512*512

typedef __attribute__((ext_vector_type(2))) float v2f;
typedef __attribute__((ext_vector_type(8))) float v8f;

__global__ __launch_bounds__(256) void gdkd_main(const float* __restrict__ ys,
                                                 const float* __restrict__ yt,
                                                 float* __restrict__ partials,
                                                 int npix)
{
    const int tid = threadIdx.x;
    const int stride = gridDim.x * blockDim.x;
    float acc = 0.0f;

    for (int p = blockIdx.x * blockDim.x + tid; p < npix; p += stride) {
        const int b    = p >> LOG2_HW;
        const int hw   = p & (HW - 1);
        const int base = ((b * CC) << LOG2_HW) + hw;

        // gfx1250 global_prefetch for the next grid-stride iteration
        {
            const int pn = p + stride;
            if (pn < npix) {
                const int basen = (((pn >> LOG2_HW) * CC) << LOG2_HW) + (pn & (HW - 1));
                __builtin_prefetch(&yt[basen], 0, 0);
                __builtin_prefetch(&ys[basen], 0, 0);
            }
        }

        float t[CC], s[CC], et[CC];
        #pragma unroll
        for (int c = 0; c < CC; ++c) t[c] = yt[base + (c << LOG2_HW)];
        #pragma unroll
        for (int c = 0; c < CC; ++c) s[c] = ys[base + (c << LOG2_HW)];

        // teacher top-3 (v1 >= v2 >= v3); mask1 = (t[c] >= v3)
        float v1 = -INFINITY, v2 = -INFINITY, v3 = -INFINITY;
        #pragma unroll
        for (int c = 0; c < CC; ++c) {
            const float x = t[c];
            if (x > v1)      { v3 = v2; v2 = v1; v1 = x; }
            else if (x > v2) { v3 = v2; v2 = x; }
            else if (x > v3) { v3 = x; }
        }
        const float maxt = v1;

        float maxs = s[0];
        #pragma unroll
        for (int c = 1; c < CC; ++c) maxs = fmaxf(maxs, s[c]);

        // one set of exponentials per tensor; all four softmax variants share them
        float Tt = 0.f, Ot = 0.f, Ts = 0.f, Os = 0.f;
        #pragma unroll
        for (int c = 0; c < CC; ++c) {
            const bool top = (t[c] >= v3);
            const float e  = __expf(t[c] - maxt);
            et[c] = e;
            const float es = __expf(s[c] - maxs);
            if (top) { Tt += e; Ts += es; }
            else     { Ot += e; Os += es; }
        }
        const float St = Tt + Ot;
        const float Ss = Ts + Os;

        const float lSt = __logf(St), lSs = __logf(Ss);
        const float lTt = __logf(Tt), lTs = __logf(Ts);
        const float lOt = __logf(Ot), lOs = __logf(Os);

        // high loss: binary KL over (top-k mass, rest mass)
        const float invSt = 1.f / St;
        const float high = (Tt * invSt) * ((lTt - lSt) - (lTs - lSs))
                         + (Ot * invSt) * ((lOt - lSt) - (lOs - lSs));

        // restricted KLs: log q = (logit - max) - log(subset sum); masked-out
        // classes contribute exactly 0 in fp32 (exp underflow), matching ref.
        float lt = 0.f, lo = 0.f;
        #pragma unroll
        for (int c = 0; c < CC; ++c) {
            const float d = (t[c] - maxt) - (s[c] - maxs);
            if (t[c] >= v3) lt += et[c] * (d - lTt + lTs);
            else            lo += et[c] * (d - lOt + lOs);
        }
        lt /= Tt;
        lo /= Ot;

        acc += high + lt + 2.0f * lo;   // W0=1, W1=1, W2=2 ; T^2=1
    }

    __shared__ float red[256];
    red[tid] = acc;
    __syncthreads();
    #pragma unroll
    for (int off = 128; off > 0; off >>= 1) {
        if (tid < off) red[tid] += red[tid + off];
        __syncthreads();
    }
    if (tid == 0) partials[blockIdx.x] = red[0];
}

// Final reduction: one wave32. Lane partials are packed into a 16x4 f32 A
// matrix (A[m][0]=s_m, A[m][2]=s_{m+16}, other K = 0) and multiplied by an
// all-ones B with v_wmma_f32_16x16x4_f32: D[m][n] = s_m + s_{m+16}. Summing
// the 8 D VGPRs per lane + shfl_xor(16) gives the exact f32 total.
__global__ void gdkd_reduce(const float* __restrict__ partials, int n,
                            float* __restrict__ out, float scale)
{
    const int lane = threadIdx.x;  // blockDim.x == 32, EXEC all ones
    float s = 0.f;
    for (int i = lane; i < n; i += 32) s += partials[i];

    v2f a;  a.x = s;   a.y = 0.f;
    v2f bb; bb.x = 1.f; bb.y = 1.f;
    v8f c = {};
    v8f d = __builtin_amdgcn_wmma_f32_16x16x4_f32(
        /*neg_a=*/false, a, /*neg_b=*/false, bb,
        /*c_mod=*/(short)0, c, /*reuse_a=*/false, /*reuse_b=*/false);

    float r = d[0] + d[1] + d[2] + d[3] + d[4] + d[5] + d[6] + d[7];
    const float total = r + __shfl_xor(r, 16, 32);
    if (lane == 0) out[0] = total * scale;
}

extern "C" void kernel_launch(void* const* d_in, const int* in_sizes, int n_in,
                              void* d_out, int out_size, void* d_ws, size_t ws_size,
                              hipStream_t stream) {
    const float* ys = (const float*)d_in[0];
    const float* yt = (const float*)d_in[1];
    float* out = (float*)d_out;
    float* partials = (float*)d_ws;

    const int npix = in_sizes[0] / CC;   // 2,097,152

    int nblocks = 2048;
    if ((size_t)nblocks * sizeof(float) > ws_size)
        nblocks = (int)(ws_size / sizeof(float));

    gdkd_main<<<nblocks, 256, 0, stream>>>(ys, yt, partials, npix);
    gdkd_reduce<<<1, 32, 0, stream>>>(partials, nblocks, out, 1.0f / (float)npix);
}